// AutoEncoder_36490042147230
// MI455X (gfx1250) — compile-verified
//
#include <hip/hip_runtime.h>

// Problem sizes (fixed by the reference)
#define DIMS 64
#define NC   400
#define NW   4000
#define KCH  10            // k-split of the w dimension
#define CHUNK (NW / KCH)   // 400 words per chunk
#define RSTRIDE 402        // LDS row stride (floats): 402*4 % 8 == 0, bank-conflict free

typedef __attribute__((ext_vector_type(2))) float v2f;
typedef __attribute__((ext_vector_type(8))) float v8f;

// ---------------------------------------------------------------------------
// Kernel 1: fused sparse gather + WMMA GEMM.
//   P[d,w]  = E[d, cid[w], w] * word_embeddings[w,d]   (only nonzero of E col)
//   partial[kch][d,c] = sum_{w in chunk} P[d,w] * (cid[w]==c)
// One workgroup = 16 d-rows x one 400-word k-chunk; 25 waves = 25 c-tiles.
// ---------------------------------------------------------------------------
__global__ void k_stage_gemm(const float* __restrict__ we,   // [NW,DIMS]
                             const float* __restrict__ E,    // [DIMS,NC,NW]
                             const int*   __restrict__ cid,  // [NW]
                             float*       __restrict__ partial) // [KCH,DIMS,NC]
{
    __shared__ float Plds[16 * RSTRIDE];
    __shared__ int   cidlds[CHUNK];

    const int dtile = blockIdx.x;        // 0..3   -> d rows [16*dtile, +16)
    const int kch   = blockIdx.y;        // 0..KCH-1
    const int t     = threadIdx.x;       // 0..799 (25 waves)
    const int w0    = kch * CHUNK;

    if (t < CHUNK) cidlds[t] = cid[w0 + t];

    // Stage P chunk: the E gather (random 4B within a contiguous 16KB row) is
    // the dominant memory cost of the whole problem (~33MB of cachelines).
    for (int i = t; i < 16 * CHUNK; i += 800) {
        const int r  = i / CHUNK;
        const int tw = i - r * CHUNK;
        const int w  = w0 + tw;
        const int d  = dtile * 16 + r;
        const int c  = cid[w];
        const float e = E[((size_t)(d * NC + c)) * (size_t)NW + (size_t)w];
        Plds[r * RSTRIDE + tw] = e * we[w * DIMS + d];
    }
    __syncthreads();

    const int wave  = t >> 5;            // 0..24 -> c-tile
    const int lane  = t & 31;
    const int n     = lane & 15;         // N (class) / M (dim-row) position
    const int khalf = lane >> 4;         // which K pair this lane carries
    const int cg    = wave * 16 + n;     // this lane's class column

    v8f acc = {};
    for (int k = 0; k < CHUNK; k += 4) {
        const int j = k + 2 * khalf;
        // A 16x4 f32: lane m holds P[d_m, w0+j], P[d_m, w0+j+1]
        v2f a = *(const v2f*)&Plds[n * RSTRIDE + j];
        // B 4x16 f32: indicator column built from class_ids (same K packing as A)
        int2 cc = *(const int2*)&cidlds[j];
        v2f b;
        b.x = (cc.x == cg) ? 1.0f : 0.0f;
        b.y = (cc.y == cg) ? 1.0f : 0.0f;
        // D = A*B + C  (8 args: neg_a, A, neg_b, B, c_mod, C, reuse_a, reuse_b)
        acc = __builtin_amdgcn_wmma_f32_16x16x4_f32(
            false, a, false, b, (short)0, acc, false, false);
    }

    // C/D layout: VGPR v, lanes 0-15 -> M=v, lanes 16-31 -> M=v+8; N=lane&15
    float* out = partial + (size_t)kch * (DIMS * NC);
#pragma unroll
    for (int v = 0; v < 8; ++v) {
        const int m = v + 8 * khalf;
        out[(dtile * 16 + m) * NC + cg] = acc[v];
    }
}

// ---------------------------------------------------------------------------
// Kernel 2: deterministic k-chunk reduction (ascending w order) -> C output.
// ---------------------------------------------------------------------------
__global__ void k_reduceC(const float* __restrict__ partial,
                          float*       __restrict__ Cout) // [DIMS,NC]
{
    const int idx = blockIdx.x * blockDim.x + threadIdx.x; // 0..25599
    float s = 0.0f;
#pragma unroll
    for (int j = 0; j < KCH; ++j) s += partial[j * (DIMS * NC) + idx];
    Cout[idx] = s;
}

// ---------------------------------------------------------------------------
// Kernel 3: W_out[d,w] = D[d,w,cid[w]] * C[d,cid[w]]  (D's column is 1-sparse)
// ---------------------------------------------------------------------------
__global__ void k_wout(const float* __restrict__ Dm,   // [DIMS,NW,NC]
                       const int*   __restrict__ cid,  // [NW]
                       const float* __restrict__ Cin,  // [DIMS,NC]
                       float*       __restrict__ Wout) // [DIMS,NW]
{
    const int idx = blockIdx.x * blockDim.x + threadIdx.x; // 0..255999
    const int d = idx / NW;
    const int w = idx - d * NW;
    const int c = cid[w];
    const float dv = Dm[((size_t)(d * NW + w)) * (size_t)NC + (size_t)c];
    Wout[idx] = dv * Cin[d * NC + c];
}

extern "C" void kernel_launch(void* const* d_in, const int* in_sizes, int n_in,
                              void* d_out, int out_size, void* d_ws, size_t ws_size,
                              hipStream_t stream) {
    const float* we  = (const float*)d_in[0];  // word_embeddings [NW,DIMS]
    const float* E   = (const float*)d_in[1];  // [DIMS,NC,NW]
    const float* Dm  = (const float*)d_in[2];  // [DIMS,NW,NC]
    const int*   cid = (const int*)  d_in[3];  // [NW]

    float* Wout = (float*)d_out;               // first output: [DIMS,NW]
    float* Cout = (float*)d_out + DIMS * NW;   // second output: [DIMS,NC]

    float* partial = (float*)d_ws;             // [KCH,DIMS,NC] = 1.0 MB scratch

    k_stage_gemm<<<dim3(DIMS / 16, KCH), 800, 0, stream>>>(we, E, cid, partial);
    k_reduceC<<<dim3((DIMS * NC) / 256), 256, 0, stream>>>(partial, Cout);
    k_wout<<<dim3((DIMS * NW) / 256), 256, 0, stream>>>(Dm, cid, Cout, Wout);
}